// Sampler1D_37383395344605
// MI455X (gfx1250) — compile-verified
//
#include <hip/hip_runtime.h>

// 1D linear texture fetch: out[u, :] = lerp(tex[i0, :], tex[i1, :], w(u))
// Memory-bound: ~276 MB of HBM traffic -> ~12 us floor at 23.3 TB/s.
//  - texture (4 MB) stays resident in the 192 MB L2 (each row reused ~64x)
//  - output (256 MB) streams via non-temporal b128 stores (won't evict texture)
//  - coords are DMA-staged to LDS via CDNA5 global_load_async_to_lds_b32
//    (ASYNCcnt), barrier-free since each lane stages its own samples
//  - uniform full-block fast path: no per-sample guards -> no exec-mask
//    save/restore, async loads keep the saddr+voffset scale_offset form
//  - all offsets 32-bit so addressing stays SGPR-base + 32-bit VGPR offset

typedef float f32x4 __attribute__((ext_vector_type(4)));

typedef __attribute__((address_space(1))) int gas_int;  // global
typedef __attribute__((address_space(3))) int las_int;  // LDS

#define TPB 256           // 8 wave32 per block
#define SPT 4             // samples per thread
#define CHUNK (TPB * SPT) // 1024 samples per block

#if __has_builtin(__builtin_amdgcn_global_load_async_to_lds_b32) && \
    __has_builtin(__builtin_amdgcn_s_wait_asynccnt)
#define USE_ASYNC_LDS 1
#else
#define USE_ASYNC_LDS 0
#endif

__device__ __forceinline__ void sample_one(
    const f32x4* __restrict__ tex4, float p, unsigned u,
    float* __restrict__ out, int nTexM1, float nM1f)
{
    // t = p*(N-1); i0 = clip(floor(t), 0, N-1); i1 = clip(i0+1); w = t - i0
    const float t  = p * nM1f;
    float fi = floorf(t);
    fi = fminf(fmaxf(fi, 0.0f), nM1f);
    const int   i0 = (int)fi;
    const int   i1 = (i0 < nTexM1) ? (i0 + 1) : nTexM1;
    const float w  = t - fi;
    const float ow = 1.0f - w;

    // Gather two 64 B rows (L2-resident texture): 8x global_load_b128.
    const unsigned r0 = (unsigned)i0 * 4u;
    const unsigned r1 = (unsigned)i1 * 4u;
    f32x4 a0 = tex4[r0 + 0], a1 = tex4[r0 + 1], a2 = tex4[r0 + 2], a3 = tex4[r0 + 3];
    f32x4 b0 = tex4[r1 + 0], b1 = tex4[r1 + 1], b2 = tex4[r1 + 2], b3 = tex4[r1 + 3];

    f32x4 o0 = a0 * ow + b0 * w;
    f32x4 o1 = a1 * ow + b1 * w;
    f32x4 o2 = a2 * ow + b2 * w;
    f32x4 o3 = a3 * ow + b3 * w;

    // Streaming 256 MB output: non-temporal b128 stores.
    f32x4* __restrict__ orow = (f32x4*)out + (u * 4u);  // u*16 floats, 32-bit
    __builtin_nontemporal_store(o0, orow + 0);
    __builtin_nontemporal_store(o1, orow + 1);
    __builtin_nontemporal_store(o2, orow + 2);
    __builtin_nontemporal_store(o3, orow + 3);
}

__global__ __launch_bounds__(TPB) void sampler1d_lerp_kernel(
    const float* __restrict__ tex,    // (65536, 16) f32, row-major
    const float* __restrict__ param,  // (U,) f32 in [0,1]
    float* __restrict__ out,          // (U, 16) f32
    int U, int nTexM1)
{
    const unsigned tid  = threadIdx.x;
    const unsigned base = blockIdx.x * CHUNK;
    const bool     full = (base + CHUNK) <= (unsigned)U;  // uniform per block

#if USE_ASYNC_LDS
    __shared__ float sp[CHUNK];
    // Async DMA this block's coordinate chunk into LDS (tracked by ASYNCcnt).
    // Each lane stages its own SPT coords -> no barrier needed, just the wait.
    if (full) {
        #pragma unroll
        for (int j = 0; j < SPT; ++j) {
            const unsigned k = (unsigned)j * TPB + tid;
            __builtin_amdgcn_global_load_async_to_lds_b32(
                (gas_int*)(param + (base + k)),
                (las_int*)(&sp[k]), /*offset=*/0, /*cpol=*/0);
        }
    } else {
        #pragma unroll
        for (int j = 0; j < SPT; ++j) {
            const unsigned k = (unsigned)j * TPB + tid;
            if (base + k < (unsigned)U) {
                __builtin_amdgcn_global_load_async_to_lds_b32(
                    (gas_int*)(param + (base + k)),
                    (las_int*)(&sp[k]), /*offset=*/0, /*cpol=*/0);
            }
        }
    }
    __builtin_amdgcn_s_wait_asynccnt(0);
#endif

    const f32x4* __restrict__ tex4 = (const f32x4*)tex;
    const float nM1f = (float)nTexM1;

    if (full) {
        #pragma unroll
        for (int j = 0; j < SPT; ++j) {
            const unsigned k = (unsigned)j * TPB + tid;
#if USE_ASYNC_LDS
            const float p = sp[k];
#else
            const float p = __builtin_nontemporal_load(param + (base + k));
#endif
            sample_one(tex4, p, base + k, out, nTexM1, nM1f);
        }
    } else {
        #pragma unroll
        for (int j = 0; j < SPT; ++j) {
            const unsigned k = (unsigned)j * TPB + tid;
            if (base + k >= (unsigned)U) return;  // k monotone in j
#if USE_ASYNC_LDS
            const float p = sp[k];
#else
            const float p = __builtin_nontemporal_load(param + (base + k));
#endif
            sample_one(tex4, p, base + k, out, nTexM1, nM1f);
        }
    }
}

extern "C" void kernel_launch(void* const* d_in, const int* in_sizes, int n_in,
                              void* d_out, int out_size, void* d_ws, size_t ws_size,
                              hipStream_t stream) {
    const float* tex   = (const float*)d_in[0];  // (65536*16,)
    const float* param = (const float*)d_in[1];  // (U,)
    float*       out   = (float*)d_out;          // (U*16,)

    const int C      = 16;
    const int nTex   = in_sizes[0] / C;          // 65536
    const int U      = in_sizes[1];              // 4194304
    const int blocks = (U + CHUNK - 1) / CHUNK;  // 4096

    sampler1d_lerp_kernel<<<dim3(blocks), dim3(TPB), 0, stream>>>(
        tex, param, out, U, nTex - 1);
}